// MeanTopKPooling2D_51960514347657
// MI455X (gfx1250) — compile-verified
//
#include <hip/hip_runtime.h>

// MeanTopKPooling2D for x:[16,112,112,128] f32, k=3, top4 -> out:[16,110,110,128]
// Faithful to the reference's buggy reshape: for output pixel r=y2*110+x2 and
// candidate q: t=q*12100+r, p=t%9, u=t/9, Y=u/110+p/3, X=u%110+p%3, value=x[b,Y,X,c].
// Channel index passes through, so each candidate is a contiguous 128-ch (512B) vector.

namespace {

constexpr int NH = 110, NW = 110;
constexpr int NPIX = 16 * NH * NW;   // 193600 output pixels
constexpr int WAVES = 4;             // waves per block (wave32)
constexpr int BLOCKS = 6050;         // 24200 waves -> 8 pixels per wave

__device__ __forceinline__ float4 f4max(float4 a, float4 b) {
  return make_float4(fmaxf(a.x, b.x), fmaxf(a.y, b.y), fmaxf(a.z, b.z), fmaxf(a.w, b.w));
}
__device__ __forceinline__ float4 f4min(float4 a, float4 b) {
  return make_float4(fminf(a.x, b.x), fminf(a.y, b.y), fminf(a.z, b.z), fminf(a.w, b.w));
}
__device__ __forceinline__ void cas(float4 &hi, float4 &lo) {
  float4 h = f4max(hi, lo);
  float4 l = f4min(hi, lo);
  hi = h; lo = l;
}

// element offset (in floats) of candidate q for pixel (b, r=y2*110+x2).
// Called with q = lane: all nine offsets computed in ONE vectorized pass.
__device__ __forceinline__ int src_elem_q(int b, int r, int q) {
  int t = q * 12100 + r;
  int p = t % 9;
  int u = t / 9;
  int Y = u / 110 + p / 3;
  int X = u % 110 + p % 3;
  return ((b * 112 + Y) * 112 + X) * 128;
}

__global__ __launch_bounds__(WAVES * 32)
void meantopk_kernel(const float* __restrict__ x, float* __restrict__ out) {
  const int lane   = threadIdx.x & 31;
  const int wid    = threadIdx.x >> 5;
  const int gw     = blockIdx.x * WAVES + wid;   // global wave id = pixel cursor
  const int stride = gridDim.x * WAVES;

  // per-wave private double buffer: [WAVES][2 bufs][9 q][32 lanes] float4 = 36 KB
  __shared__ float4 sm[WAVES * 2 * 9 * 32];

  const unsigned lane16 = (unsigned)lane << 4;                  // reused VADDR (bytes)
  const unsigned smbase = (unsigned)(unsigned long long)&sm[0];
  const unsigned mybase = smbase + (unsigned)(wid * 2 * 9 * 32 * 16) + lane16;

  // Issue 9 async 512B gathers for one pixel into LDS buffer `buf`.
  // Lane q computes candidate q's offset (one ~22-instr VALU pass for all nine),
  // v_readlane broadcasts each to an SGPR, which rides in the async-load SADDR.
  auto issue = [&](int pix, int buf) {
    int pixu = __builtin_amdgcn_readfirstlane(pix);
    int b = pixu / 12100;
    int r = pixu - b * 12100;
    int voff = src_elem_q(b, r, lane);             // lanes 0..8 meaningful
    unsigned lbase = mybase + (unsigned)(buf * 9 * 512);
#pragma unroll
    for (int q = 0; q < 9; ++q) {
      int offq = __builtin_amdgcn_readlane(voff, q);  // uniform -> SGPR
      const float* src = x + offq;                    // scalar 64-bit add
      unsigned laddr = lbase + (unsigned)(q * 512);   // one v_add per q
      asm volatile("global_load_async_to_lds_b128 %0, %1, %2"
                   :
                   : "v"(laddr), "v"(lane16), "s"(src)
                   : "memory");
    }
  };

  // Top-4-of-9 mean per channel via bitonic select:
  //   sort a=v0..3, b=v4..7 descending; L_i = max(a_i, b_{3-i}) is the top-4 of 8;
  //   top4of9 sum = (sum(L)+v8) - min(L, v8).
  auto compute_store = [&](int pix, int buf) {
    const float4* sl = &sm[((wid * 2 + buf) * 9) * 32 + lane];
    float4 a0 = sl[0 * 32], a1 = sl[1 * 32], a2 = sl[2 * 32], a3 = sl[3 * 32];
    cas(a0, a1); cas(a2, a3); cas(a0, a2); cas(a1, a3); cas(a1, a2);
    float4 b0 = sl[4 * 32], b1 = sl[5 * 32], b2 = sl[6 * 32], b3 = sl[7 * 32];
    cas(b0, b1); cas(b2, b3); cas(b0, b2); cas(b1, b3); cas(b1, b2);
    float4 v8 = sl[8 * 32];
    float4 L0 = f4max(a0, b3), L1 = f4max(a1, b2);
    float4 L2 = f4max(a2, b1), L3 = f4max(a3, b0);
    float4 m = f4min(f4min(L0, L1), f4min(L2, v8));
    m = f4min(m, L3);
    float4 res = make_float4(
        (L0.x + L1.x + L2.x + L3.x + v8.x - m.x) * 0.25f,
        (L0.y + L1.y + L2.y + L3.y + v8.y - m.y) * 0.25f,
        (L0.z + L1.z + L2.z + L3.z + v8.z - m.z) * 0.25f,
        (L0.w + L1.w + L2.w + L3.w + v8.w - m.w) * 0.25f);
    reinterpret_cast<float4*>(out)[(unsigned)(pix * 32 + lane)] = res;
  };

  int pix = gw;
  int buf = 0;
  if (pix < NPIX) issue(pix, 0);
  for (; pix < NPIX; pix += stride, buf ^= 1) {
    int nxt = pix + stride;
    if (nxt < NPIX) {
      issue(nxt, buf ^ 1);                              // prefetch next tile
      asm volatile("s_wait_asynccnt 9" ::: "memory");   // current tile landed (in-order)
    } else {
      asm volatile("s_wait_asynccnt 0" ::: "memory");
    }
    compute_store(pix, buf);
  }
}

}  // namespace

extern "C" void kernel_launch(void* const* d_in, const int* in_sizes, int n_in,
                              void* d_out, int out_size, void* d_ws, size_t ws_size,
                              hipStream_t stream) {
  (void)in_sizes; (void)n_in; (void)out_size; (void)d_ws; (void)ws_size;
  const float* x = reinterpret_cast<const float*>(d_in[0]);
  float* out = reinterpret_cast<float*>(d_out);
  meantopk_kernel<<<dim3(BLOCKS), dim3(WAVES * 32), 0, stream>>>(x, out);
}